// FisherKolmogorovPDE_39376260169920
// MI455X (gfx1250) — compile-verified
//
#include <hip/hip_runtime.h>
#include <stdint.h>

// Fisher-Kolmogorov PDE: out = div(D * grad(u)) + rho*u*(1-u)
// Grid: [B=4, 1, 192, 192, 192] fp32, replicate-boundary central differences.
// Bandwidth-bound (~453 MB min traffic -> ~19.5us @ 23.3 TB/s). Strategy:
// z-marching blocks with LDS plane ring buffers filled via CDNA5 async
// global->LDS copies (ASYNCcnt), one plane ahead, so u/D are read from HBM
// exactly once and reused from LDS. rho/out are streamed with non-temporal
// hints so they don't evict the u/D halo planes that neighbor blocks re-hit
// in the 192MB L2.

#define N   192
#define N2  (N * N)
#define N3  (N * N * N)

#define TX  32
#define TY  8
#define NTHREADS (TX * TY)
#define ZC  48
#define NCHUNK (N / ZC)   // 4

// u needs halo 2 (x,y), D needs halo 1
#define WU (TX + 4)       // 36
#define HU (TY + 4)       // 12
#define PU (WU * HU)      // 432 floats / u plane
#define WD (TX + 2)       // 34
#define HD (TY + 2)       // 10
#define PD (WD * HD)      // 340 floats / D plane

#define URING 6
#define DRING 4

typedef __attribute__((address_space(1))) int gas_int;
typedef __attribute__((address_space(3))) int las_int;

__device__ __forceinline__ int clampN(int v) {
  return v < 0 ? 0 : (v > N - 1 ? N - 1 : v);
}

// --- CDNA5 async global -> LDS copy (tracked by ASYNCcnt) ------------------
__device__ __forceinline__ void async_copy_f32(const float* g, float* l) {
#if __has_builtin(__builtin_amdgcn_global_load_async_to_lds_b32)
  __builtin_amdgcn_global_load_async_to_lds_b32(
      (gas_int*)g, (las_int*)l, /*imm offset*/ 0, /*cpol*/ 0);
#else
  asm volatile("global_load_async_to_lds_b32 %0, %1, off"
               :
               : "v"((uint32_t)(uintptr_t)l), "v"(g)
               : "memory");
#endif
}

__device__ __forceinline__ void wait_async0() {
#if __has_builtin(__builtin_amdgcn_s_wait_asynccnt)
  __builtin_amdgcn_s_wait_asynccnt(0);
#else
  asm volatile("s_wait_asynccnt 0" ::: "memory");
#endif
}

// --- plane loaders (clamped halos baked in) --------------------------------
__device__ __forceinline__ void load_u_plane(const float* __restrict__ ub, float* su,
                                             int slot, int zq, int x0, int y0, int tid) {
  const float* src = ub + (size_t)clampN(zq) * N2;
  float* dst = su + slot * PU;
#pragma unroll
  for (int e0 = 0; e0 < PU; e0 += NTHREADS) {
    int e = e0 + tid;
    if (e < PU) {
      int hy = e / WU;
      int hx = e - hy * WU;
      int gy = clampN(y0 - 2 + hy);
      int gx = clampN(x0 - 2 + hx);
      async_copy_f32(src + gy * N + gx, dst + e);
    }
  }
}

__device__ __forceinline__ void load_d_plane(const float* __restrict__ Db, float* sD,
                                             int slot, int zq, int x0, int y0, int tid) {
  const float* src = Db + (size_t)clampN(zq) * N2;
  float* dst = sD + slot * PD;
#pragma unroll
  for (int e0 = 0; e0 < PD; e0 += NTHREADS) {
    int e = e0 + tid;
    if (e < PD) {
      int hy = e / WD;
      int hx = e - hy * WD;
      int gy = clampN(y0 - 1 + hy);
      int gx = clampN(x0 - 1 + hx);
      async_copy_f32(src + gy * N + gx, dst + e);
    }
  }
}

// ---------------------------------------------------------------------------
__global__ __launch_bounds__(NTHREADS)
void fk_pde_kernel(const float* __restrict__ u,
                   const float* __restrict__ D,
                   const float* __restrict__ rho,
                   float* __restrict__ out) {
  __shared__ float su[URING * PU];   // 6 * 432 * 4B = 10368 B
  __shared__ float sD[DRING * PD];   // 4 * 340 * 4B =  5440 B

  const int tx = threadIdx.x;
  const int ty = threadIdx.y;
  const int tid = ty * TX + tx;

  const int x0 = blockIdx.x * TX;
  const int y0 = blockIdx.y * TY;
  const int bz = blockIdx.z;
  const int batch = bz / NCHUNK;
  const int z0 = (bz - batch * NCHUNK) * ZC;

  const size_t base = (size_t)batch * N3;
  const float* ub = u + base;
  const float* Db = D + base;

  // ---- preload ring: u planes z0-2..z0+2, D planes z0-1..z0+1 -------------
  for (int zq = z0 - 2; zq <= z0 + 2; ++zq)
    load_u_plane(ub, su, (zq + 2 * URING) % URING, zq, x0, y0, tid);
  for (int zq = z0 - 1; zq <= z0 + 1; ++zq)
    load_d_plane(Db, sD, (zq + 2 * DRING) % DRING, zq, x0, y0, tid);
  wait_async0();
  __syncthreads();

  const int gx = x0 + tx;
  const int gy = y0 + ty;
  const int ux = tx + 2, uy = ty + 2;   // LDS coords (halo 2)
  const int dxc = tx + 1, dyc = ty + 1; // LDS coords (halo 1)

  // boundary-dependent stencil offsets (derived from the double clamped
  // central difference; interior = {+2,0,0,-2})
  const int xa = (gx == N - 1) ? 0 : 2;
  const int xb = (gx == N - 1) ? -1 : 0;
  const int xc = (gx == 0) ? 1 : 0;
  const int xd = (gx == 0) ? 0 : -2;
  const int ya = (gy == N - 1) ? 0 : 2;
  const int yb = (gy == N - 1) ? -1 : 0;
  const int yc = (gy == 0) ? 1 : 0;
  const int yd = (gy == 0) ? 0 : -2;

#define UL(slot, ddx, ddy) su[(slot) * PU + (uy + (ddy)) * WU + (ux + (ddx))]
#define DLd(slot, ddx, ddy) sD[(slot) * PD + (dyc + (ddy)) * WD + (dxc + (ddx))]

  for (int z = z0; z < z0 + ZC; ++z) {
    // software pipeline: fetch plane z+3 (u) and z+2 (D) while computing z
    load_u_plane(ub, su, (z + 3) % URING, z + 3, x0, y0, tid);
    load_d_plane(Db, sD, (z + 2) % DRING, z + 2, x0, y0, tid);

    // uniform ring slots (SALU)
    const int za = (z == N - 1) ? 0 : 2;
    const int zb = (z == N - 1) ? -1 : 0;
    const int zc_ = (z == 0) ? 1 : 0;
    const int zd = (z == 0) ? 0 : -2;

    const int s_c  = (z + 2 * URING) % URING;        // dz = 0
    const int s_za = (z + za + 2 * URING) % URING;
    const int s_zb = (z + zb + 2 * URING) % URING;
    const int s_zc = (z + zc_ + 2 * URING) % URING;
    const int s_zd = (z + zd + 2 * URING) % URING;

    const int t_m = (z - 1 + 2 * DRING) % DRING;     // dz = -1
    const int t_c = (z + 2 * DRING) % DRING;         // dz =  0
    const int t_p = (z + 1 + 2 * DRING) % DRING;     // dz = +1

    const float uc = UL(s_c, 0, 0);

    const float termx =
        DLd(t_c, 1, 0)  * (UL(s_c, xa, 0) - UL(s_c, xb, 0)) -
        DLd(t_c, -1, 0) * (UL(s_c, xc, 0) - UL(s_c, xd, 0));

    const float termy =
        DLd(t_c, 0, 1)  * (UL(s_c, 0, ya) - UL(s_c, 0, yb)) -
        DLd(t_c, 0, -1) * (UL(s_c, 0, yc) - UL(s_c, 0, yd));

    const float termz =
        DLd(t_p, 0, 0) * (UL(s_za, 0, 0) - UL(s_zb, 0, 0)) -
        DLd(t_m, 0, 0) * (UL(s_zc, 0, 0) - UL(s_zd, 0, 0));

    const size_t idx = base + (size_t)z * N2 + (size_t)gy * N + gx;
    // rho: single-use stream; out: write-once stream -> non-temporal so they
    // don't evict u/D halo planes from L2 (cross-block reuse).
    const float r = __builtin_nontemporal_load(&rho[idx]);
    const float val = 0.25f * (termx + termy + termz) + r * uc * (1.0f - uc);
    __builtin_nontemporal_store(val, &out[idx]);

    wait_async0();
    __syncthreads();
  }
#undef UL
#undef DLd
}

extern "C" void kernel_launch(void* const* d_in, const int* in_sizes, int n_in,
                              void* d_out, int out_size, void* d_ws, size_t ws_size,
                              hipStream_t stream) {
  (void)n_in; (void)d_ws; (void)ws_size; (void)out_size;
  const float* u   = (const float*)d_in[0];
  const float* D   = (const float*)d_in[1];
  const float* rho = (const float*)d_in[2];
  float* out = (float*)d_out;

  const int B = in_sizes[0] / N3;   // batch (=4 for the reference shapes)

  dim3 block(TX, TY, 1);
  dim3 grid(N / TX, N / TY, B * NCHUNK);
  fk_pde_kernel<<<grid, block, 0, stream>>>(u, D, rho, out);
}